// KalmanFilter_84688165142669
// MI455X (gfx1250) — compile-verified
//
#include <hip/hip_runtime.h>
#include <math.h>

// CDNA5 wave32 fp32 WMMA: D(16x16) = A(16x4) * B(4x16) + C
typedef __attribute__((ext_vector_type(2))) float v2f;
typedef __attribute__((ext_vector_type(8))) float v8f;

#define SDIM 32
#define ODIM 16
#define LOG2PI 1.8378770664093453f

__device__ __forceinline__ float mld(const float* Mp, int r, int c, int ld, bool tr) {
  return tr ? Mp[c * ld + r] : Mp[r * ld + c];
}

// Computes D[mo:mo+16, no:no+16] = sgnA * opA(A) * opB(B) + Cin (Cin==nullptr -> 0)
// All matrices row-major in LDS. Kd must be a multiple of 4. Wave-scoped (lane = tid&31).
__device__ __forceinline__ void wmma_tile16(
    const float* A, int lda, bool tA, float sgnA,
    const float* B, int ldb, bool tB,
    const float* Cin, int ldc,
    float* D, int ldd,
    int mo, int no, int Kd, int lane)
{
  const int half = lane >> 4;   // 0: lanes 0-15, 1: lanes 16-31
  const int l15  = lane & 15;
  v8f acc;
  if (Cin) {
#pragma unroll
    for (int r = 0; r < 8; ++r) acc[r] = Cin[(mo + r + half * 8) * ldc + no + l15];
  } else {
#pragma unroll
    for (int r = 0; r < 8; ++r) acc[r] = 0.0f;
  }
#pragma unroll
  for (int k0 = 0; k0 < Kd; k0 += 4) {
    const int ka = k0 + half * 2;           // ISA A/B fragment layout for 16x16x4 f32
    v2f a, b;
    a[0] = sgnA * mld(A, mo + l15, ka + 0, lda, tA);
    a[1] = sgnA * mld(A, mo + l15, ka + 1, lda, tA);
    b[0] = mld(B, ka + 0, no + l15, ldb, tB);
    b[1] = mld(B, ka + 1, no + l15, ldb, tB);
    acc = __builtin_amdgcn_wmma_f32_16x16x4_f32(false, a, false, b, (short)0, acc,
                                                false, false);
  }
#pragma unroll
  for (int r = 0; r < 8; ++r) D[(mo + r + half * 8) * ldd + no + l15] = acc[r];
}

__global__ __launch_bounds__(128)
void kalman_filter_kernel(const float* __restrict__ y,  const float* __restrict__ Ag,
                          const float* __restrict__ Cg, const float* __restrict__ Qg,
                          const float* __restrict__ Rg, const float* __restrict__ z0g,
                          const float* __restrict__ P0g, float* __restrict__ out, int T)
{
  const int N = SDIM, M = ODIM;
  __shared__ float sA[SDIM * SDIM], sQ[SDIM * SDIM], sI[SDIM * SDIM];
  __shared__ float sC[ODIM * SDIM], sR[ODIM * ODIM];
  __shared__ float sPa[SDIM * SDIM], sPb[SDIM * SDIM];
  __shared__ float sT1[SDIM * SDIM], sPpred[SDIM * SDIM];
  __shared__ float sIKC[SDIM * SDIM], sT2[SDIM * SDIM];
  __shared__ float sCP[ODIM * SDIM], sPCt[SDIM * ODIM];
  __shared__ float sK[SDIM * ODIM], sKR[SDIM * ODIM];
  __shared__ float sS[ODIM * ODIM], sSinv[ODIM * ODIM], sLinv[ODIM * ODIM];
  __shared__ float sza[SDIM], szb[SDIM], szp[SDIM];
  __shared__ float sinnov[ODIM], sv[ODIM], slog[ODIM];

  const int tid  = (int)threadIdx.x;
  const int lane = tid & 31;
  const int wave = tid >> 5;

  float* outStates = out;
  float* outCov    = out + (size_t)T * N;
  float* outLL     = out + (size_t)T * N + (size_t)T * N * N;

  // ---- one-time staging into LDS + t=0 outputs ----
  for (int i = tid; i < N * N; i += 128) {
    sA[i]  = Ag[i];
    sQ[i]  = Qg[i];
    sI[i]  = ((i >> 5) == (i & 31)) ? 1.0f : 0.0f;
    sPa[i] = P0g[i];
    outCov[i] = P0g[i];
  }
  for (int i = tid; i < M * N; i += 128) sC[i] = Cg[i];
  for (int i = tid; i < M * M; i += 128) sR[i] = Rg[i];
  if (tid < N) { sza[tid] = z0g[tid]; outStates[tid] = z0g[tid]; }
  __syncthreads();

  float* Pc = sPa; float* Pn = sPb;   // ping-pong covariance
  float* zc = sza; float* zn = szb;   // ping-pong state
  float ll = 0.0f;                    // only tid==0's copy matters

  const int mo = (wave >> 1) * 16;    // this wave's default 16x16 tile
  const int no = (wave & 1) * 16;

  for (int t = 1; t < T; ++t) {
    const float* yt = y + (size_t)t * M;

    // z_pred = A z  (wave 0)
    if (tid < N) {
      float s = 0.f;
      for (int j = 0; j < N; ++j) s += sA[tid * N + j] * zc[j];
      szp[tid] = s;
    }
    // T1 = A * P   (4 tiles, one per wave)
    wmma_tile16(sA, N, false, 1.f, Pc, N, false, nullptr, 0, sT1, N, mo, no, N, lane);
    __syncthreads();

    // Ppred = T1 * A^T + Q
    wmma_tile16(sT1, N, false, 1.f, sA, N, true, sQ, N, sPpred, N, mo, no, N, lane);
    __syncthreads();

    // innov = y_t - C z_pred ; CP = C*Ppred (waves 0,1) ; PCt = Ppred*C^T (waves 2,3)
    if (tid < M) {
      float s = 0.f;
      for (int j = 0; j < N; ++j) s += sC[tid * N + j] * szp[j];
      sinnov[tid] = yt[tid] - s;
    }
    if (wave == 0) wmma_tile16(sC, N, false, 1.f, sPpred, N, false, nullptr, 0, sCP, N, 0,  0, N, lane);
    if (wave == 1) wmma_tile16(sC, N, false, 1.f, sPpred, N, false, nullptr, 0, sCP, N, 0, 16, N, lane);
    if (wave == 2) wmma_tile16(sPpred, N, false, 1.f, sC, N, true, nullptr, 0, sPCt, M, 0,  0, N, lane);
    if (wave == 3) wmma_tile16(sPpred, N, false, 1.f, sC, N, true, nullptr, 0, sPCt, M, 16, 0, N, lane);
    __syncthreads();

    // S = CP * C^T + R  (wave 0)
    if (wave == 0) wmma_tile16(sCP, N, false, 1.f, sC, N, true, sR, M, sS, M, 0, 0, N, lane);
    __syncthreads();

    // Cholesky S=LL^T (in place, lower), L^{-1}, Sinv = L^{-T}L^{-1}, quad, ll update.
    // Entirely within wave 0: per-wave in-order LDS, no barriers needed inside.
    if (wave == 0) {
      for (int j = 0; j < M; ++j) {
        float d = sqrtf(sS[j * M + j]);
        float Lij = 0.f;
        if (lane > j && lane < M) { Lij = sS[lane * M + j] / d; sS[lane * M + j] = Lij; }
        if (lane == j) { sS[j * M + j] = d; slog[j] = logf(d); }
        if (lane > j && lane < M) {
          for (int k = j + 1; k <= lane; ++k) sS[lane * M + k] -= Lij * sS[k * M + j];
        }
      }
      if (lane < M) {            // forward-substitute column `lane` of L^{-1}
        const int j = lane;
        for (int i = 0; i < j; ++i) sLinv[i * M + j] = 0.f;
        sLinv[j * M + j] = 1.0f / sS[j * M + j];
        for (int i = j + 1; i < M; ++i) {
          float s = 0.f;
          for (int k = j; k < i; ++k) s += sS[i * M + k] * sLinv[k * M + j];
          sLinv[i * M + j] = -s / sS[i * M + i];
        }
      }
      // Sinv = Linv^T * Linv (one WMMA tile, K=16)
      wmma_tile16(sLinv, M, true, 1.f, sLinv, M, false, nullptr, 0, sSinv, M, 0, 0, M, lane);
      if (lane < M) {
        float s = 0.f;
        for (int j = 0; j < M; ++j) s += sSinv[lane * M + j] * sinnov[j];
        sv[lane] = s;
      }
      if (lane == 0) {
        float quad = 0.f, sl = 0.f;
        for (int i = 0; i < M; ++i) { quad += sinnov[i] * sv[i]; sl += slog[i]; }
        ll += -0.5f * (2.0f * sl + quad + (float)M * LOG2PI);
      }
    }
    __syncthreads();

    // K = PCt * Sinv  (waves 0,1)
    if (wave == 0) wmma_tile16(sPCt, M, false, 1.f, sSinv, M, false, nullptr, 0, sK, M, 0,  0, M, lane);
    if (wave == 1) wmma_tile16(sPCt, M, false, 1.f, sSinv, M, false, nullptr, 0, sK, M, 16, 0, M, lane);
    __syncthreads();

    // z_new = z_pred + K innov ; IKC = I - K*C (neg folded into A-load) ; KR = K*R
    if (tid < N) {
      float s = 0.f;
      for (int j = 0; j < M; ++j) s += sK[tid * M + j] * sinnov[j];
      float z = szp[tid] + s;
      zn[tid] = z;
      outStates[(size_t)t * N + tid] = z;
    }
    wmma_tile16(sK, M, false, -1.f, sC, N, false, sI, N, sIKC, N, mo, no, M, lane);
    if (wave == 0) wmma_tile16(sK, M, false, 1.f, sR, M, false, nullptr, 0, sKR, M, 0,  0, M, lane);
    if (wave == 1) wmma_tile16(sK, M, false, 1.f, sR, M, false, nullptr, 0, sKR, M, 16, 0, M, lane);
    __syncthreads();

    // T2 = IKC * Ppred ; Pn = KR * K^T   (2 tiles per wave)
    wmma_tile16(sIKC, N, false, 1.f, sPpred, N, false, nullptr, 0, sT2, N, mo, no, N, lane);
    wmma_tile16(sKR, M, false, 1.f, sK, M, true, nullptr, 0, Pn, N, mo, no, M, lane);
    __syncthreads();

    // Pn += T2 * IKC^T   (Joseph form complete)
    wmma_tile16(sT2, N, false, 1.f, sIKC, N, true, Pn, N, Pn, N, mo, no, N, lane);
    __syncthreads();

    // stream covariance out, swap ping-pong buffers
    {
      float* dst = outCov + (size_t)t * (N * N);
      for (int i = tid; i < N * N; i += 128) dst[i] = Pn[i];
    }
    { float* tp = Pc; Pc = Pn; Pn = tp; }
    { float* tz = zc; zc = zn; zn = tz; }
    __syncthreads();
  }

  if (tid == 0) *outLL = ll;
}

extern "C" void kernel_launch(void* const* d_in, const int* in_sizes, int n_in,
                              void* d_out, int out_size, void* d_ws, size_t ws_size,
                              hipStream_t stream) {
  (void)n_in; (void)out_size; (void)d_ws; (void)ws_size;
  const float* y  = (const float*)d_in[0];
  const float* A  = (const float*)d_in[1];
  const float* C  = (const float*)d_in[2];
  const float* Q  = (const float*)d_in[3];
  const float* R  = (const float*)d_in[4];
  const float* z0 = (const float*)d_in[5];
  const float* P0 = (const float*)d_in[6];
  float* out = (float*)d_out;
  const int T = in_sizes[0] / ODIM;
  hipLaunchKernelGGL(kalman_filter_kernel, dim3(1), dim3(128), 0, stream,
                     y, A, C, Q, R, z0, P0, out, T);
}